// GAT_68839735820520
// MI455X (gfx1250) — compile-verified
//
#include <hip/hip_runtime.h>

#define N_NODES 50000
#define N_EDGES 800000
#define E_TOT   (N_EDGES + N_NODES)   // with self-loops
#define IN_C    128
#define HID_C   32
#define OUT_C   64
#define HEADS   8
#define HC1     (HEADS * HID_C)       // 256
#define KC      128                   // K-chunk staged in LDS (32 KB)

typedef __attribute__((ext_vector_type(2))) float v2f;
typedef __attribute__((ext_vector_type(8))) float v8f;

// ---- monotonic float <-> uint encoding for atomicMax-based segment max ----
__device__ __forceinline__ unsigned f2ord(float f) {
  unsigned u = __float_as_uint(f);
  return (u & 0x80000000u) ? ~u : (u | 0x80000000u);
}
__device__ __forceinline__ float ord2f(unsigned u) {
  unsigned v = (u & 0x80000000u) ? (u & 0x7FFFFFFFu) : ~u;
  return __uint_as_float(v);
}

// ---------------------------------------------------------------- zero fill
__global__ void k_zero4(float4* p, long long n4) {
  long long t = (long long)blockIdx.x * blockDim.x + threadIdx.x;
  if (t < n4) p[t] = make_float4(0.f, 0.f, 0.f, 0.f);
}

// ------------------------------------------------ fp32 WMMA GEMM: C = A @ B
// A:[M,K] row-major, B:[K,N] row-major, C:[M,N].  M%16==0, N%64==0, K%128==0.
// Block = 8 waves; each wave owns a 16x64 tile (same n-block). B is staged
// through LDS pair-interleaved so each B fragment is one ds_load_b64.
__global__ void k_gemm_wmma(const float* __restrict__ A, const float* __restrict__ B,
                            float* __restrict__ C, int M, int N, int K) {
  __shared__ float Bs[KC * 64];             // [(k/2)][n][k&1], 32 KB

  const int lane = threadIdx.x & 31;
  const int wid  = threadIdx.x >> 5;
  const int mTiles = M >> 4;
  const int nBlks  = N >> 6;  (void)nBlks;
  const int mBlk   = (mTiles + 7) >> 3;
  const int mb = blockIdx.x % mBlk;
  const int nb = blockIdx.x / mBlk;
  const int mTile  = (mb << 3) + wid;
  const bool active = (mTile < mTiles);     // wave-uniform -> EXEC all-ones
  const int m0  = mTile << 4;
  const int n0  = nb << 6;
  const int l16 = lane & 15;
  const int hi  = lane >> 4;                // K-pair half per ISA A/B layout

  v8f c0 = {}, c1 = {}, c2 = {}, c3 = {};
  const float* arow = A + (long long)(active ? (m0 + l16) : l16) * K;

  for (int kc = 0; kc < K; kc += KC) {
    // ---- cooperative stage of B[kc..kc+KC) x [n0..n0+64) ----
    for (int idx = threadIdx.x; idx < KC * 64; idx += 256) {
      const int k = idx >> 6;
      const int n = idx & 63;
      Bs[((k >> 1) << 7) + (n << 1) + (k & 1)] =
          B[(long long)(kc + k) * N + n0 + n];
    }
    __syncthreads();

    if (active) {
      float2 av = *reinterpret_cast<const float2*>(arow + kc + (hi << 1));
      for (int k0 = kc; k0 < kc + KC; k0 += 4) {
        const int kb  = k0 + (hi << 1);
        const int kbn = (kb + 4 < K) ? (kb + 4) : kb;     // clamped prefetch
        const float2 avn = *reinterpret_cast<const float2*>(arow + kbn);
        v2f a; a.x = av.x; a.y = av.y;

        const float* bp = Bs + (((k0 - kc) >> 1) + hi) * 128 + (l16 << 1);
        float2 bv; v2f b;
        bv = *reinterpret_cast<const float2*>(bp);
        b.x = bv.x; b.y = bv.y;
        c0 = __builtin_amdgcn_wmma_f32_16x16x4_f32(false, a, false, b, (short)0, c0, false, false);
        bv = *reinterpret_cast<const float2*>(bp + 32);
        b.x = bv.x; b.y = bv.y;
        c1 = __builtin_amdgcn_wmma_f32_16x16x4_f32(false, a, false, b, (short)0, c1, false, false);
        bv = *reinterpret_cast<const float2*>(bp + 64);
        b.x = bv.x; b.y = bv.y;
        c2 = __builtin_amdgcn_wmma_f32_16x16x4_f32(false, a, false, b, (short)0, c2, false, false);
        bv = *reinterpret_cast<const float2*>(bp + 96);
        b.x = bv.x; b.y = bv.y;
        c3 = __builtin_amdgcn_wmma_f32_16x16x4_f32(false, a, false, b, (short)0, c3, false, false);

        av = avn;
      }
    }
    __syncthreads();
  }

  if (active) {
    const int rbase = m0 + (hi << 3);       // C/D layout: VGPR r = rows r / r+8
    #pragma unroll
    for (int r = 0; r < 8; ++r) {
      float* crow = C + (long long)(rbase + r) * N + n0 + l16;
      crow[0]  = c0[r];
      crow[16] = c1[r];
      crow[32] = c2[r];
      crow[48] = c3[r];
    }
  }
}

// --------------------------------- per-(node,head) attention coefficients
__global__ void k_attn(const float* __restrict__ h, const float* __restrict__ asrc,
                       const float* __restrict__ adst, float* __restrict__ es,
                       float* __restrict__ ed, int n, int H, int C) {
  long long t = (long long)blockIdx.x * blockDim.x + threadIdx.x;
  if (t >= (long long)n * H) return;
  const int node = (int)(t / H), hd = (int)(t % H);
  const float* hp = h + (long long)node * H * C + (long long)hd * C;
  const float* ap = asrc + (long long)hd * C;
  const float* bp = adst + (long long)hd * C;
  float s = 0.f, d = 0.f;
  for (int c = 0; c < C; ++c) { const float v = hp[c]; s += v * ap[c]; d += v * bp[c]; }
  es[t] = s; ed[t] = d;
}

// ------------------------- edge logits: leaky_relu + segment max (atomicMax)
__global__ void k_logits(const int* __restrict__ ei, const float* __restrict__ es,
                         const float* __restrict__ ed, float* __restrict__ ee,
                         unsigned* __restrict__ menc, int H) {
  long long t = (long long)blockIdx.x * blockDim.x + threadIdx.x;
  if (t >= (long long)E_TOT * H) return;
  const int e = (int)(t / H), hd = (int)(t % H);
  const int src = (e < N_EDGES) ? ei[e] : (e - N_EDGES);
  const int dst = (e < N_EDGES) ? ei[N_EDGES + e] : (e - N_EDGES);
  float v = es[(long long)src * H + hd] + ed[(long long)dst * H + hd];
  v = (v > 0.f) ? v : 0.2f * v;
  ee[t] = v;
  atomicMax(&menc[(long long)dst * H + hd], f2ord(v));
}

// --------------------------- exp(e - max) + segment sum of exps (atomicAdd)
__global__ void k_expnorm(const int* __restrict__ ei, float* __restrict__ ee,
                          const unsigned* __restrict__ menc, float* __restrict__ den,
                          int H) {
  long long t = (long long)blockIdx.x * blockDim.x + threadIdx.x;
  if (t >= (long long)E_TOT * H) return;
  const int e = (int)(t / H), hd = (int)(t % H);
  const int dst = (e < N_EDGES) ? ei[N_EDGES + e] : (e - N_EDGES);
  const float m  = ord2f(menc[(long long)dst * H + hd]);
  const float ex = expf(ee[t] - m);
  ee[t] = ex;
  atomicAdd(&den[(long long)dst * H + hd], ex);
}

// ------------- aggregation: out[dst] += h[src]*alpha   (one wave per edge)
__global__ void k_agg(const int* __restrict__ ei, const float* __restrict__ h,
                      const float* __restrict__ ee, const float* __restrict__ den,
                      float* __restrict__ out, int H, int C) {
  const int lane = threadIdx.x & 31;
  const int e = (int)(((long long)blockIdx.x * blockDim.x + threadIdx.x) >> 5);
  if (e >= E_TOT) return;
  const int src = (e < N_EDGES) ? ei[e] : (e - N_EDGES);
  const int dst = (e < N_EDGES) ? ei[N_EDGES + e] : (e - N_EDGES);
  const int CH = H * C;
  const int chunk = CH >> 5;                 // 8 (layer1) or 2 (layer2)
  const int c0 = lane * chunk;               // chunk never crosses a head boundary
  const int hd = c0 / C;
  const float alpha = ee[(long long)e * H + hd] /
                      (den[(long long)dst * H + hd] + 1e-16f);
  const float* hs = h + (long long)src * CH + c0;
  float* op = out + (long long)dst * CH + c0;
  #pragma unroll
  for (int j = 0; j < 8; ++j)
    if (j < chunk) atomicAdd(&op[j], hs[j] * alpha);
}

// ----------------------------------------------- ELU(x + bias) elementwise
__global__ void k_elu_bias(const float* __restrict__ in, const float* __restrict__ b,
                           float* __restrict__ o, int C, long long n) {
  long long t = (long long)blockIdx.x * blockDim.x + threadIdx.x;
  if (t >= n) return;
  const float v = in[t] + b[t % C];
  o[t] = (v > 0.f) ? v : (expf(v) - 1.f);
}

// ------------------------------------------------------- bias add in place
__global__ void k_bias_inplace(float* __restrict__ o, const float* __restrict__ b,
                               int C, long long n) {
  long long t = (long long)blockIdx.x * blockDim.x + threadIdx.x;
  if (t >= n) return;
  o[t] = o[t] + b[t % C];
}

extern "C" void kernel_launch(void* const* d_in, const int* in_sizes, int n_in,
                              void* d_out, int out_size, void* d_ws, size_t ws_size,
                              hipStream_t stream) {
  const float* x   = (const float*)d_in[0];
  const int*   ei  = (const int*)d_in[1];
  const float* W1  = (const float*)d_in[2];
  const float* as1 = (const float*)d_in[3];
  const float* ad1 = (const float*)d_in[4];
  const float* b1  = (const float*)d_in[5];
  const float* W2  = (const float*)d_in[6];
  const float* as2 = (const float*)d_in[7];
  const float* ad2 = (const float*)d_in[8];
  const float* b2  = (const float*)d_in[9];
  float* out = (float*)d_out;

  // ---- carve workspace (256B-aligned slabs) ----
  char* p = (char*)d_ws;
  auto take = [&](long long elems) -> float* {
    float* r = (float*)p;
    p += ((elems * 4 + 255) / 256) * 256;
    return r;
  };
  // zero-initialized region (contiguous, zeroed once per launch)
  float*    out1 = take((long long)N_NODES * HC1);
  unsigned* m1   = (unsigned*)take((long long)N_NODES * HEADS);
  float*    den1 = take((long long)N_NODES * HEADS);
  unsigned* m2   = (unsigned*)take(N_NODES);
  float*    den2 = take(N_NODES);
  const long long zeroElems = (long long)(p - (char*)d_ws) / 4;   // multiple of 64
  // write-before-read buffers
  float* h1  = take((long long)N_NODES * HC1);   // layer-1 features; reused as h2in
  float* h2  = take((long long)N_NODES * OUT_C);
  float* es1 = take((long long)N_NODES * HEADS);
  float* ed1 = take((long long)N_NODES * HEADS);
  float* es2 = take(N_NODES);
  float* ed2 = take(N_NODES);
  float* ee  = take((long long)E_TOT * HEADS);   // edge buffer; reused for layer 2

  const dim3 blk(256);
  auto nblk = [](long long t) { return dim3((unsigned)((t + 255) / 256)); };

  // ---- init accumulators ----
  k_zero4<<<nblk(zeroElems / 4), blk, 0, stream>>>((float4*)d_ws, zeroElems / 4);
  const long long outElems = (long long)N_NODES * OUT_C;
  k_zero4<<<nblk(outElems / 4), blk, 0, stream>>>((float4*)out, outElems / 4);

  const int mBlk = ((N_NODES / 16) + 7) / 8;     // 391

  // ---- layer 1 ----
  k_gemm_wmma<<<dim3(mBlk * (HC1 / 64)), blk, 0, stream>>>(x, W1, h1, N_NODES, HC1, IN_C);
  k_attn<<<nblk((long long)N_NODES * HEADS), blk, 0, stream>>>(h1, as1, ad1, es1, ed1,
                                                               N_NODES, HEADS, HID_C);
  k_logits<<<nblk((long long)E_TOT * HEADS), blk, 0, stream>>>(ei, es1, ed1, ee, m1, HEADS);
  k_expnorm<<<nblk((long long)E_TOT * HEADS), blk, 0, stream>>>(ei, ee, m1, den1, HEADS);
  k_agg<<<nblk((long long)E_TOT * 32), blk, 0, stream>>>(ei, h1, ee, den1, out1, HEADS, HID_C);

  // ---- ELU(out1 + b1) -> h2in (aliases h1, safe: h1 fully consumed) ----
  k_elu_bias<<<nblk((long long)N_NODES * HC1), blk, 0, stream>>>(out1, b1, h1, HC1,
                                                                 (long long)N_NODES * HC1);

  // ---- layer 2 (H=1, C=64) ----
  k_gemm_wmma<<<dim3(mBlk * (OUT_C / 64)), blk, 0, stream>>>(h1, W2, h2, N_NODES, OUT_C, HC1);
  k_attn<<<nblk((long long)N_NODES), blk, 0, stream>>>(h2, as2, ad2, es2, ed2,
                                                       N_NODES, 1, OUT_C);
  k_logits<<<nblk((long long)E_TOT), blk, 0, stream>>>(ei, es2, ed2, ee, m2, 1);
  k_expnorm<<<nblk((long long)E_TOT), blk, 0, stream>>>(ei, ee, m2, den2, 1);
  k_agg<<<nblk((long long)E_TOT * 32), blk, 0, stream>>>(ei, h2, ee, den2, out, 1, OUT_C);

  // ---- out += b2 ----
  k_bias_inplace<<<nblk(outElems), blk, 0, stream>>>(out, b2, OUT_C, outElems);
}